// Gmfd_56942676410884
// MI455X (gfx1250) — compile-verified
//
#include <hip/hip_runtime.h>
#include <hip/hip_bf16.h>

// ---------------------------------------------------------------------------
// SPD projection of 4096 independent 64x64 fp32 matrices (x: [64,64,64,64]).
//   S  = 0.5*(X + X^T)
//   Y0 = S / ||S||_F                       (spectrum in [-1,1])
//   Y  <- 1.5*Y - 0.5*Y^3   (Newton-Schulz matrix-sign iteration, 12 steps)
//   |S| = S * sign(S)                      (spectral absolute value)
//   out = 0.5*(|S| + |S|^T) + EPS*mu*I     (mu ~ mean eigenvalue proxy)
// All matrix products are 64x64x64 fp32 GEMMs on V_WMMA_F32_16X16X4_F32,
// fully LDS-resident (padded stride 68 to kill bank conflicts).
// One workgroup (512 threads = 16 wave32) per matrix; one wave per 16x16 tile.
// ---------------------------------------------------------------------------

typedef __attribute__((ext_vector_type(2))) float v2f;
typedef __attribute__((ext_vector_type(8))) float v8f;

#define LDSW 68          // padded row stride in floats (16B-aligned rows)
#define EPS 1e-6f

// C(64x64) = A(64x64) * B(64x64), all in LDS with row stride LDSW.
// Wave w computes tile (w>>2, w&3). Uniform control flow: EXEC all ones.
__device__ __forceinline__ void gemm64(const float* __restrict__ A,
                                       const float* __restrict__ B,
                                       float* __restrict__ C) {
    const int tid  = threadIdx.x;
    const int wave = tid >> 5;
    const int lane = tid & 31;
    const int ti   = (wave >> 2) << 4;   // tile row base
    const int tj   = (wave & 3) << 4;    // tile col base
    const int hi   = lane >> 4;          // lane half: K +2 / row +8
    const int ln   = lane & 15;

    v8f acc = {};
#pragma unroll
    for (int kc = 0; kc < 64; kc += 4) {
        const int ka = kc + 2 * hi;
        // A fragment 16x4: lane ln holds row ti+ln, K = ka, ka+1
        v2f a;
        a.x = A[(ti + ln) * LDSW + ka];
        a.y = A[(ti + ln) * LDSW + ka + 1];
        // B fragment 4x16: lane ln holds col tj+ln, K = ka, ka+1
        v2f b;
        b.x = B[ka * LDSW + tj + ln];
        b.y = B[(ka + 1) * LDSW + tj + ln];
        // D = A*B + C   (fp32 WMMA; 8-arg form: neg_a, A, neg_b, B, c_mod, C, reuse_a, reuse_b)
        acc = __builtin_amdgcn_wmma_f32_16x16x4_f32(
            false, a, false, b, (short)0, acc, false, false);
    }
    // C/D layout: VGPR r -> row (ti + r + 8*hi), col tj+ln
#pragma unroll
    for (int r = 0; r < 8; ++r) {
        C[(ti + r + 8 * hi) * LDSW + tj + ln] = acc[r];
    }
}

__global__ __launch_bounds__(512) void spd_project_kernel(
    const float* __restrict__ x, float* __restrict__ out) {
    __shared__ float S[64 * LDSW];
    __shared__ float Y[64 * LDSW];
    __shared__ float T[64 * LDSW];
    __shared__ float U[64 * LDSW];
    __shared__ float red[2];

    const int b   = blockIdx.x;
    const int tid = threadIdx.x;
    const float* __restrict__ X = x + (size_t)b * 4096;
    float* __restrict__ O       = out + (size_t)b * 4096;

    if (tid < 2) red[tid] = 0.0f;

    // Load X (coalesced) into padded LDS buffer T.
#pragma unroll
    for (int i = 0; i < 8; ++i) {
        const int e = i * 512 + tid;
        T[(e >> 6) * LDSW + (e & 63)] = X[e];
    }
    __syncthreads();

    // S = 0.5*(X + X^T); accumulate ||S||_F^2.
    float part = 0.0f;
#pragma unroll
    for (int i = 0; i < 8; ++i) {
        const int e = i * 512 + tid;
        const int r = e >> 6, c = e & 63;
        const float v = 0.5f * (T[r * LDSW + c] + T[c * LDSW + r]);
        S[r * LDSW + c] = v;
        part += v * v;
    }
    atomicAdd(&red[0], part);
    __syncthreads();

    // Y = S / ||S||_F  (spectral radius <= 1, nonzero a.s. for Gaussian input)
    const float inv = rsqrtf(red[0] + 1e-30f);
#pragma unroll
    for (int i = 0; i < 8; ++i) {
        const int e = i * 512 + tid;
        const int idx = (e >> 6) * LDSW + (e & 63);
        Y[idx] = S[idx] * inv;
    }
    __syncthreads();

    // Newton-Schulz sign iteration: Y <- 1.5*Y - 0.5*Y^3
    for (int it = 0; it < 12; ++it) {
        gemm64(Y, Y, T);     // T = Y^2
        __syncthreads();
        gemm64(T, Y, U);     // U = Y^3
        __syncthreads();
#pragma unroll
        for (int i = 0; i < 8; ++i) {
            const int e = i * 512 + tid;
            const int idx = (e >> 6) * LDSW + (e & 63);
            Y[idx] = 1.5f * Y[idx] - 0.5f * U[idx];
        }
        __syncthreads();
    }

    // |S| = S * sign(S)
    gemm64(S, Y, T);
    __syncthreads();

    // mu = trace(|S|)/64 as eigenvalue-scale proxy for the EPS shift.
    if (tid < 64) atomicAdd(&red[1], T[tid * LDSW + tid]);
    __syncthreads();
    const float mu = red[1] * (1.0f / 64.0f);

    // out = 0.5*(|S| + |S|^T) + EPS*mu*I
#pragma unroll
    for (int i = 0; i < 8; ++i) {
        const int e = i * 512 + tid;
        const int r = e >> 6, c = e & 63;
        float v = 0.5f * (T[r * LDSW + c] + T[c * LDSW + r]);
        if (r == c) v += EPS * mu;
        O[e] = v;
    }
}

extern "C" void kernel_launch(void* const* d_in, const int* in_sizes, int n_in,
                              void* d_out, int out_size, void* d_ws, size_t ws_size,
                              hipStream_t stream) {
    (void)n_in; (void)d_ws; (void)ws_size; (void)out_size;
    const float* x = (const float*)d_in[0];
    float* out     = (float*)d_out;
    const int n_mat = in_sizes[0] / 4096;   // 64*64 = 4096 matrices of 64x64
    spd_project_kernel<<<n_mat, 512, 0, stream>>>(x, out);
}